// GATConvModel_33354716021175
// MI455X (gfx1250) — compile-verified
//
#include <hip/hip_runtime.h>
#include <hip/hip_bf16.h>
#include <math.h>

#define N_NODES   50000
#define N_EDGES   800000
#define IN_FEATS  128
#define OUT_FEATS 16
#define NUM_HEADS 4
#define HD        (NUM_HEADS * OUT_FEATS)   // 64
#define NEG_SLOPE 0.2f

typedef __attribute__((ext_vector_type(2))) float v2f;
typedef __attribute__((ext_vector_type(8))) float v8f;

// ---------------------------------------------------------------------------
// Kernel 1: feat = X @ W   ([50000,128] x [128,64]) via V_WMMA_F32_16X16X4_F32
// One wave32 computes one 16x16 output tile; 4 waves/block cover all 64 cols
// of a 16-row stripe. K loop: 128 / 4 = 32 WMMA issues per wave.
// A layout (16x4 f32): lanes 0-15 -> row M=lane, a={K+0,K+1};
//                      lanes 16-31 -> row M=lane-16, a={K+2,K+3}.
// B layout (4x16 f32): symmetric transpose of A (col N=lane&15).
// C/D layout: VGPR v, lanes 0-15 -> (M=v, N=lane); lanes 16-31 -> (M=v+8).
// ---------------------------------------------------------------------------
__global__ __launch_bounds__(128) void gat_gemm_wmma(
    const float* __restrict__ X, const float* __restrict__ W,
    float* __restrict__ feat) {
  const int lane  = threadIdx.x & 31;
  const int wave  = threadIdx.x >> 5;       // tileN index: 0..3
  const int tileM = blockIdx.x;             // 0..3124
  const int l15   = lane & 15;
  const int hi    = lane >> 4;              // 0 for lanes 0-15, 1 for 16-31

  const int mrow  = tileM * 16 + l15;       // A row this lane feeds
  const int ncol  = (wave << 4) + l15;      // B/C column this lane feeds
  const int koff  = hi << 1;                // 0 or 2

  const float* xptr = X + (size_t)mrow * IN_FEATS + koff;
  const float* wptr = W + (size_t)koff * HD + ncol;

  v8f c = {};
#pragma unroll 4
  for (int k = 0; k < IN_FEATS; k += 4) {
    v2f a = *(const v2f*)xptr;              // K+0,K+1 (or K+2,K+3)
    v2f b;
    b.x = wptr[0];                          // W[kbase  ][ncol]
    b.y = wptr[HD];                         // W[kbase+1][ncol]
    c = __builtin_amdgcn_wmma_f32_16x16x4_f32(
        /*neg_a=*/false, a, /*neg_b=*/false, b,
        /*c_mod=*/(short)0, c, /*reuse_a=*/false, /*reuse_b=*/false);
    xptr += 4;
    wptr += 4 * HD;
  }

  const int mbase = tileM * 16 + (hi << 3); // rows v.. / v+8..
  float* op = feat + (size_t)mbase * HD + ncol;
#pragma unroll
  for (int v = 0; v < 8; ++v) op[(size_t)v * HD] = c[v];
}

// ---------------------------------------------------------------------------
// Kernel 2: el[n,h] = feat[n,h,:] . attn_l[h,:] ;  er likewise
// ---------------------------------------------------------------------------
__global__ void gat_node_attn(const float* __restrict__ feat,
                              const float* __restrict__ attn_l,
                              const float* __restrict__ attn_r,
                              float* __restrict__ el, float* __restrict__ er) {
  int idx = blockIdx.x * blockDim.x + threadIdx.x;  // n*H + h
  if (idx >= N_NODES * NUM_HEADS) return;
  int h = idx & (NUM_HEADS - 1);
  int n = idx >> 2;
  const float* f  = feat + (size_t)n * HD + h * OUT_FEATS;
  const float* al = attn_l + h * OUT_FEATS;
  const float* ar = attn_r + h * OUT_FEATS;
  float sl = 0.f, sr = 0.f;
#pragma unroll
  for (int d = 0; d < OUT_FEATS; ++d) {
    float v = f[d];
    sl += v * al[d];
    sr += v * ar[d];
  }
  el[idx] = sl;
  er[idx] = sr;
}

// ---------------------------------------------------------------------------
// Kernel 3: init out=bias, m=-inf, denom=0
// ---------------------------------------------------------------------------
__global__ void gat_init(float* __restrict__ out, const float* __restrict__ bias,
                         float* __restrict__ m, float* __restrict__ denom) {
  int idx = blockIdx.x * blockDim.x + threadIdx.x;
  if (idx >= N_NODES * HD) return;
  int j = idx & (HD - 1);
  out[idx] = bias[j];
  if (j < NUM_HEADS) {
    int n = idx >> 6;
    m[n * NUM_HEADS + j]     = -__builtin_huge_valf();
    denom[n * NUM_HEADS + j] = 0.f;
  }
}

// float atomic-max via signed/unsigned int ordering trick
__device__ __forceinline__ void atomicMaxF(float* addr, float val) {
  if (val >= 0.f)
    atomicMax((int*)addr, __float_as_int(val));
  else
    atomicMin((unsigned int*)addr, (unsigned int)__float_as_int(val));
}

// ---------------------------------------------------------------------------
// Kernel 4: e = leaky_relu(el[src]+er[dst]); segment max into m[dst]
// ---------------------------------------------------------------------------
__global__ void gat_edge_score(const int* __restrict__ src,
                               const int* __restrict__ dst,
                               const float* __restrict__ el,
                               const float* __restrict__ er,
                               float* __restrict__ esc, float* __restrict__ m) {
  int e = blockIdx.x * blockDim.x + threadIdx.x;
  if (e >= N_EDGES) return;
  int s = src[e], d = dst[e];
#pragma unroll
  for (int h = 0; h < NUM_HEADS; ++h) {
    float v = el[s * NUM_HEADS + h] + er[d * NUM_HEADS + h];
    v = v > 0.f ? v : NEG_SLOPE * v;
    esc[e * NUM_HEADS + h] = v;
    atomicMaxF(&m[d * NUM_HEADS + h], v);
  }
}

// ---------------------------------------------------------------------------
// Kernel 5: e_exp = exp(e - m[dst]); segment sum into denom[dst]
// ---------------------------------------------------------------------------
__global__ void gat_edge_exp(const int* __restrict__ dst,
                             float* __restrict__ esc,
                             const float* __restrict__ m,
                             float* __restrict__ denom) {
  int idx = blockIdx.x * blockDim.x + threadIdx.x;  // e*H + h
  if (idx >= N_EDGES * NUM_HEADS) return;
  int e = idx >> 2;
  int h = idx & (NUM_HEADS - 1);
  int d = dst[e];
  float v = expf(esc[idx] - m[d * NUM_HEADS + h]);
  esc[idx] = v;
  atomicAdd(&denom[d * NUM_HEADS + h], v);
}

// ---------------------------------------------------------------------------
// Kernel 6: out[dst, j] += (e_exp / denom[dst]) * feat[src, j]
// 256 threads/block = 4 edges/block, 64 feature lanes per edge.
// feat (12.8 MB) is L2-resident, so the src gathers hit L2.
// ---------------------------------------------------------------------------
__global__ __launch_bounds__(256) void gat_aggregate(
    const int* __restrict__ src, const int* __restrict__ dst,
    const float* __restrict__ esc, const float* __restrict__ denom,
    const float* __restrict__ feat, float* __restrict__ out) {
  int t = threadIdx.x;
  int e = blockIdx.x * 4 + (t >> 6);
  if (e >= N_EDGES) return;
  int j = t & (HD - 1);
  int h = j >> 4;
  int s = src[e], d = dst[e];
  float den = denom[d * NUM_HEADS + h];
  float alpha = esc[e * NUM_HEADS + h] / (den == 0.f ? 1.f : den);
  atomicAdd(&out[(size_t)d * HD + j], alpha * feat[(size_t)s * HD + j]);
}

// ---------------------------------------------------------------------------
// Launch
// ---------------------------------------------------------------------------
extern "C" void kernel_launch(void* const* d_in, const int* in_sizes, int n_in,
                              void* d_out, int out_size, void* d_ws, size_t ws_size,
                              hipStream_t stream) {
  const float* inputs  = (const float*)d_in[0];
  const int*   src     = (const int*)d_in[1];
  const int*   dst     = (const int*)d_in[2];
  const float* W       = (const float*)d_in[3];
  const float* attn_l  = (const float*)d_in[4];
  const float* attn_r  = (const float*)d_in[5];
  const float* bias    = (const float*)d_in[6];
  float* out = (float*)d_out;

  // workspace layout (floats)
  float* ws    = (float*)d_ws;
  float* feat  = ws;                                 // N*64
  float* el    = feat + (size_t)N_NODES * HD;        // N*4
  float* er    = el + (size_t)N_NODES * NUM_HEADS;   // N*4
  float* m     = er + (size_t)N_NODES * NUM_HEADS;   // N*4
  float* denom = m + (size_t)N_NODES * NUM_HEADS;    // N*4
  float* esc   = denom + (size_t)N_NODES * NUM_HEADS;// E*4

  // 1) GEMM via WMMA: 3125 blocks of 4 waves (one 16-row stripe each)
  gat_gemm_wmma<<<N_NODES / 16, 128, 0, stream>>>(inputs, W, feat);

  // 2) per-node attention halves
  {
    int n = N_NODES * NUM_HEADS;
    gat_node_attn<<<(n + 255) / 256, 256, 0, stream>>>(feat, attn_l, attn_r, el, er);
  }

  // 3) init out = bias, m = -inf, denom = 0
  {
    int n = N_NODES * HD;
    gat_init<<<(n + 255) / 256, 256, 0, stream>>>(out, bias, m, denom);
  }

  // 4) edge scores + segment max
  gat_edge_score<<<(N_EDGES + 255) / 256, 256, 0, stream>>>(src, dst, el, er, esc, m);

  // 5) exp + segment sum
  {
    int n = N_EDGES * NUM_HEADS;
    gat_edge_exp<<<(n + 255) / 256, 256, 0, stream>>>(dst, esc, m, denom);
  }

  // 6) weighted scatter aggregate (bias already in out)
  gat_aggregate<<<(N_EDGES + 3) / 4, 256, 0, stream>>>(src, dst, esc, denom, feat, out);
}